// FlowCPABLinear_22952305229987
// MI455X (gfx1250) — compile-verified
//
#include <hip/hip_runtime.h>
#include <math.h>

// ---------------------------------------------------------------------------
// CPAB 1D flow (NC=5 cells), forward transform + log|d phi/dx|.
// Compute-bound elementwise kernel: ~300 VALU ops/pt incl. exp/log per step.
// Strategy: wave-uniform early exit of the 7-step scan, hw transcendentals,
// async global->LDS double-buffered streaming (gfx1250 ASYNCcnt path).
// cpab_main is defined FIRST so the disasm snippet shows its pipeline.
// ---------------------------------------------------------------------------

#define NCELL 5
#define BLOCK 256
#define PPT 4
#define TILE_PTS (BLOCK * PPT)      // 1024 points per tile per block
#define MAX_STEPS (NCELL + 2)
#define GRID_BLOCKS 4096

#if defined(__gfx1250__)
  #if __has_builtin(__builtin_amdgcn_global_load_async_to_lds_b128)
    #define HAVE_ASYNC_LDS 1
  #else
    #define HAVE_ASYNC_LDS 0
  #endif
  #if __has_builtin(__builtin_amdgcn_s_wait_asynccnt)
    #define WAIT_ASYNC(n) do { __builtin_amdgcn_s_wait_asynccnt(n); asm volatile("" ::: "memory"); } while (0)
  #else
    #define WAIT_ASYNC(n) asm volatile("s_wait_asynccnt %0" :: "i"(n) : "memory")
  #endif
  #if __has_builtin(__builtin_amdgcn_ballot_w32)
    #define ALL_DONE(p) (__builtin_amdgcn_ballot_w32(!(p)) == 0u)
  #else
    #define ALL_DONE(p) (0)
  #endif
#else
  #define HAVE_ASYNC_LDS 0
  #define WAIT_ASYNC(n) do { } while (0)
  #define ALL_DONE(p) (0)
#endif

#if HAVE_ASYNC_LDS
// Builtin signature (from clang diagnostics): (v4i addrspace(1)*, v4i addrspace(3)*, imm, imm)
typedef int v4i_t __attribute__((ext_vector_type(4)));
__device__ __forceinline__ void async_load16(const float* gsrc, float* ldst)
{
    __builtin_amdgcn_global_load_async_to_lds_b128(
        (__attribute__((address_space(1))) v4i_t*)(gsrc),
        (__attribute__((address_space(3))) v4i_t*)(ldst),
        0, 0);
}
#endif

// ---------------------------------------------------------------------------
// One point: closed-form CPA integration, exact mirror of the JAX scan step,
// with a wave-uniform early exit once every lane has finished.
// ---------------------------------------------------------------------------
__device__ __forceinline__ void cpab_point(float xin, const float* As,
                                           float a, float b, float c, float d,
                                           float lca, float& zz, float& lz)
{
    const float ncf = (float)NCELL;
    float xc = fminf(fmaxf(xin, 1e-7f), 1.0f - 1e-7f);
    xc = fmaf(a, xc, b);
    float t = 1.0f;
    float logg = 0.0f;
    bool done = false;
    #pragma unroll 1
    for (int s = 0; s < MAX_STEPS; ++s) {
        if (ALL_DONE(done)) break;
        int cell = (int)floorf(xc * ncf);
        cell = cell < 0 ? 0 : (cell > NCELL - 1 ? NCELL - 1 : cell);
        float2 ab = *(const float2*)(As + 2 * cell);        // ds_load_b64
        float ac = ab.x, bc = ab.y;
        float v = fmaf(ac, xc, bc);
        bool az = fabsf(ac) < 1e-8f;
        float sa = az ? 1.0f : ac;
        float r = bc / sa;
        float e = __expf(ac * t);                           // v_exp_f32
        float psi = az ? fmaf(bc, t, xc) : fmaf(xc + r, e, -r);
        float xl = (float)cell * (1.0f / ncf);
        float xr = xl + (1.0f / ncf);
        float xb = (v >= 0.0f) ? xr : xl;
        float den = xc + r;
        den = (fabsf(den) < 1e-12f) ? 1e-12f : den;
        float frac = az ? 1.0f : (xb + r) / den;
        frac = fmaxf(frac, 1e-12f);
        float sb = (fabsf(bc) < 1e-12f) ? 1e-12f : bc;
        float t_hit = az ? (xb - xc) / sb : (__logf(frac) / sa);  // v_log_f32
        bool stays = (psi >= xl) && (psi <= xr);
        bool exits = ((v < 0.0f) && (cell == 0)) || ((v >= 0.0f) && (cell == NCELL - 1));
        bool finish = stays || exits;
        float dt = finish ? t : t_hit;
        if (!done) {
            logg = fmaf(ac, dt, logg);
            xc = finish ? psi : xb;
            t  = finish ? 0.0f : (t - t_hit);
            done = finish;
        }
    }
    zz = fmaf(c, xc, d);
    lz = lca + logg;
}

__device__ __forceinline__ void compute_store4(const float* __restrict__ x,
                                               float4 xv, long long base, int N,
                                               const float* As, float a, float b,
                                               float c, float d, float lca,
                                               float* __restrict__ out)
{
    float xin[PPT] = {xv.x, xv.y, xv.z, xv.w};
    if (base + PPT > (long long)N) {            // rare tail: guarded scalar fetch
        #pragma unroll
        for (int k = 0; k < PPT; ++k)
            xin[k] = (base + k < (long long)N) ? x[base + k] : 0.5f;
    }
    float zs[PPT], ls[PPT];
    #pragma unroll
    for (int k = 0; k < PPT; ++k)
        cpab_point(xin[k], As, a, b, c, d, lca, zs[k], ls[k]);
    float* outZ = out;
    float* outL = out + N;
    if (base + PPT <= (long long)N) {
        *(float4*)(outZ + base) = make_float4(zs[0], zs[1], zs[2], zs[3]);  // global_store_b128
        *(float4*)(outL + base) = make_float4(ls[0], ls[1], ls[2], ls[3]);
    } else {
        for (int k = 0; k < PPT; ++k)
            if (base + k < (long long)N) { outZ[base + k] = zs[k]; outL[base + k] = ls[k]; }
    }
}

// ---------------------------------------------------------------------------
// Main kernel: double-buffered async global->LDS streaming. Each thread
// async-loads its own 16B slice; each wave waits only on its own ASYNCcnt,
// so no barriers are needed in the pipeline.
// ---------------------------------------------------------------------------
__global__ __launch_bounds__(BLOCK) void cpab_main(const float* __restrict__ x,
                                                   const float* __restrict__ ws,
                                                   float* __restrict__ out,
                                                   int N, int nTiles)
{
    __shared__ __align__(16) float As[16];
    const int tid = threadIdx.x;
    if (tid < 16) As[tid] = ws[tid];
    __syncthreads();
    const float a = As[10], b = As[11], c = As[12], d = As[13], lca = As[14];

#if HAVE_ASYNC_LDS
    __shared__ __align__(16) float xbuf[2][TILE_PTS];
    long long tile = blockIdx.x;
    if (tile < (long long)nTiles) {
        long long lb0 = tile * TILE_PTS + tid * PPT;
        if (lb0 + PPT > N) lb0 = (long long)N - PPT;
        if (lb0 < 0) lb0 = 0;
        async_load16(x + lb0, &xbuf[0][tid * PPT]);   // global_load_async_to_lds_b128
        int buf = 0;
        while (true) {
            long long next = tile + (long long)gridDim.x;
            bool haveNext = next < (long long)nTiles;
            if (haveNext) {
                long long bn = next * TILE_PTS + tid * PPT;
                if (bn + PPT > N) bn = (long long)N - PPT;
                if (bn < 0) bn = 0;
                async_load16(x + bn, &xbuf[buf ^ 1][tid * PPT]);
                WAIT_ASYNC(1);     // oldest (current buf) complete
            } else {
                WAIT_ASYNC(0);
            }
            float4 xv = *(const float4*)(&xbuf[buf][tid * PPT]);   // ds_load_b128
            compute_store4(x, xv, tile * TILE_PTS + tid * PPT, N, As, a, b, c, d, lca, out);
            if (!haveNext) break;
            tile = next;
            buf ^= 1;
        }
    }
#else
    for (long long tile = blockIdx.x; tile < (long long)nTiles; tile += gridDim.x) {
        long long base = tile * TILE_PTS + tid * PPT;
        long long lb = base;
        if (lb + PPT > N) lb = (long long)N - PPT;
        if (lb < 0) lb = 0;
        __builtin_prefetch(x + lb + (long long)gridDim.x * TILE_PTS, 0, 3);  // global_prefetch_b8
        float4 xv = *(const float4*)(x + lb);
        compute_store4(x, xv, base, N, As, a, b, c, d, lca, out);
    }
#endif
}

// ---------------------------------------------------------------------------
// Setup: build orthonormal null-space basis of the 1D CPA continuity
// constraints (spanned by hat functions; MGS in f64), fold theta into per-cell
// [slope, offset] pairs, stash scalars + log(c*a) in workspace.
// ---------------------------------------------------------------------------
__global__ void cpab_setup(const float* __restrict__ theta,
                           const float* __restrict__ pa, const float* __restrict__ pb,
                           const float* __restrict__ pc, const float* __restrict__ pd,
                           float* __restrict__ ws)
{
    if (blockIdx.x != 0 || threadIdx.x != 0) return;
    const int nc = NCELL;
    double Hb[NCELL + 1][2 * NCELL];
    // Hat-function j (value 1 at knot j) -> interleaved (slope, offset) coeffs.
    for (int j = 0; j <= nc; ++j) {
        double h[NCELL + 1];
        for (int m = 0; m <= nc; ++m) h[m] = (m == j) ? 1.0 : 0.0;
        for (int i = 0; i < nc; ++i) {
            double slope = (h[i + 1] - h[i]) * (double)nc;
            double off   = h[i] - slope * ((double)i / (double)nc);
            Hb[j][2 * i]     = slope;
            Hb[j][2 * i + 1] = off;
        }
    }
    // Modified Gram-Schmidt -> orthonormal basis of the same null space.
    for (int j = 0; j <= nc; ++j) {
        for (int k = 0; k < j; ++k) {
            double dot = 0.0;
            for (int m = 0; m < 2 * nc; ++m) dot += Hb[j][m] * Hb[k][m];
            for (int m = 0; m < 2 * nc; ++m) Hb[j][m] -= dot * Hb[k][m];
        }
        double nrm = 0.0;
        for (int m = 0; m < 2 * nc; ++m) nrm += Hb[j][m] * Hb[j][m];
        nrm = sqrt(nrm);
        for (int m = 0; m < 2 * nc; ++m) Hb[j][m] /= nrm;
    }
    // A = B @ theta  (columns of B are the basis vectors)
    for (int m = 0; m < 2 * nc; ++m) {
        double s = 0.0;
        for (int j = 0; j <= nc; ++j) s += Hb[j][m] * (double)theta[j];
        ws[m] = (float)s;
    }
    float a = pa[0], b = pb[0], c = pc[0], d = pd[0];
    ws[10] = a; ws[11] = b; ws[12] = c; ws[13] = d;
    ws[14] = logf(c * a);   // log(c*exp(g)*a) == log(c*a) + g
    ws[15] = 0.0f;
}

// ---------------------------------------------------------------------------
// Launcher. Inputs: x(N), theta(6), a(1), b(1), c(1), d(1).
// Output: [z(N) ; log_dz_dx(N)] concatenated (out_size == 2N).
// ---------------------------------------------------------------------------
extern "C" void kernel_launch(void* const* d_in, const int* in_sizes, int n_in,
                              void* d_out, int out_size, void* d_ws, size_t ws_size,
                              hipStream_t stream)
{
    const float* x     = (const float*)d_in[0];
    const float* theta = (const float*)d_in[1];
    const float* pa    = (const float*)d_in[2];
    const float* pb    = (const float*)d_in[3];
    const float* pc    = (const float*)d_in[4];
    const float* pd    = (const float*)d_in[5];
    float* ws  = (float*)d_ws;
    float* out = (float*)d_out;
    const int N = in_sizes[0];

    cpab_setup<<<1, 32, 0, stream>>>(theta, pa, pb, pc, pd, ws);

    int nTiles = (N + TILE_PTS - 1) / TILE_PTS;
    int blocks = nTiles < GRID_BLOCKS ? nTiles : GRID_BLOCKS;
    if (blocks < 1) blocks = 1;
    cpab_main<<<blocks, BLOCK, 0, stream>>>(x, ws, out, N, nTiles);
}